// ScaledDotProductAttentionPoolVariableKSTDDual_19035295056358
// MI455X (gfx1250) — compile-verified
//
#include <hip/hip_runtime.h>
#include <math.h>

#define B_N   128
#define L_N   1024
#define D_N   64
#define LOUT  1022           // 1 cls + 1021 kept
#define NKEEP 1021
#define SSTR  1028           // padded LDS row stride (floats); 4112B rows stay 16B-aligned

typedef __attribute__((ext_vector_type(2))) float v2f;
typedef __attribute__((ext_vector_type(4))) float v4f;
typedef __attribute__((ext_vector_type(8))) float v8f;
typedef __attribute__((ext_vector_type(4))) int   v4i;

__device__ __forceinline__ float wave_max32(float v) {
#pragma unroll
  for (int o = 16; o > 0; o >>= 1) v = fmaxf(v, __shfl_xor(v, o, 32));
  return v;
}
__device__ __forceinline__ float wave_sum32(float v) {
#pragma unroll
  for (int o = 16; o > 0; o >>= 1) v += __shfl_xor(v, o, 32);
  return v;
}

// ---------------------------------------------------------------------------
// Kernel 1: S = QK^T / 8 via V_WMMA_F32_16X16X4_F32, row softmax, row std.
// Grid: (L/16, B), 256 threads (8 waves). Each WG = 16-query strip x 1024 keys.
// ---------------------------------------------------------------------------
__global__ __launch_bounds__(256) void k_qk_softmax_std(
    const float* __restrict__ q, const float* __restrict__ kmat,
    float* __restrict__ attn, float* __restrict__ stdt)
{
  __shared__ float ss[16 * SSTR];          // 16 x 1024 score strip (padded)
  const int b    = blockIdx.y;
  const int qt   = blockIdx.x;
  const int tid  = threadIdx.x;
  const int w    = tid >> 5;
  const int lane = tid & 31;

  const int m  = lane & 15;                // WMMA row/col index for this lane
  const int k0 = (lane < 16) ? 0 : 2;      // 32-bit A/B frag K-pair base

  // A fragments: Q strip rows, all 16 K-chunks of D=64 (register-resident)
  const float* qp = q + ((size_t)b * L_N + (size_t)(qt * 16 + m)) * D_N;
  v2f afrag[16];
#pragma unroll
  for (int c = 0; c < 16; ++c) {
    afrag[c].x = qp[c * 4 + k0];
    afrag[c].y = qp[c * 4 + k0 + 1];
  }

  const float* kb = kmat + (size_t)b * L_N * D_N;
  for (int kt = w; kt < 64; kt += 8) {     // 8 key-tiles per wave
    const float* kp = kb + (size_t)(kt * 16 + m) * D_N;  // B frag col n = m
    if (kt + 8 < 64)
      __builtin_prefetch(kp + 8 * 16 * D_N, 0, 1);       // next tile for this wave
    v8f acc = {};
#pragma unroll
    for (int c = 0; c < 16; ++c) {         // K = 64 in steps of 4
      v2f bfrag;
      bfrag.x = kp[c * 4 + k0];
      bfrag.y = kp[c * 4 + k0 + 1];
      acc = __builtin_amdgcn_wmma_f32_16x16x4_f32(false, afrag[c], false, bfrag,
                                                  (short)0, acc, false, false);
    }
    const int rbase = (lane < 16) ? 0 : 8; // C/D layout: VGPR v -> row v (+8 for hi lanes)
    const int col   = kt * 16 + m;
#pragma unroll
    for (int vv = 0; vv < 8; ++vv)
      ss[(vv + rbase) * SSTR + col] = acc[vv] * 0.125f;   // 1/sqrt(64)
  }
  __syncthreads();

  // Row softmax + std(ddof=1); 8 waves x 2 rows; float4-wide streaming
#pragma unroll
  for (int rr = 0; rr < 2; ++rr) {
    const int r = w * 2 + rr;
    float* row  = &ss[r * SSTR];
    float mx = -3.402823e38f;
    for (int j = lane * 4; j < L_N; j += 128) {
      v4f x = *(const v4f*)(row + j);
      mx = fmaxf(mx, fmaxf(fmaxf(x.x, x.y), fmaxf(x.z, x.w)));
    }
    mx = wave_max32(mx);
    float s = 0.f;
    for (int j = lane * 4; j < L_N; j += 128) {
      v4f x = *(const v4f*)(row + j);
      v4f e;
      e.x = __expf(x.x - mx); e.y = __expf(x.y - mx);
      e.z = __expf(x.z - mx); e.w = __expf(x.w - mx);
      *(v4f*)(row + j) = e;
      s += (e.x + e.y) + (e.z + e.w);
    }
    s = wave_sum32(s);
    const float inv = 1.0f / s;
    float* arow = attn + ((size_t)b * L_N + (size_t)(qt * 16 + r)) * L_N;
    float s2 = 0.f;
    for (int j = lane * 4; j < L_N; j += 128) {
      v4f e = *(const v4f*)(row + j);
      v4f p = e * inv;
      *(v4f*)(arow + j) = p;
      s2 += (p.x * p.x + p.y * p.y) + (p.z * p.z + p.w * p.w);
    }
    s2 = wave_sum32(s2);
    const int qr = qt * 16 + r;
    if (lane == 0 && qr >= 1) {            // std over attn_t rows only
      const float mean = 1.0f / (float)L_N;          // row sums to 1
      float var = (s2 - (float)L_N * mean * mean) * (1.0f / (float)(L_N - 1));
      stdt[(size_t)b * (L_N - 1) + (qr - 1)] = sqrtf(fmaxf(var, 0.f));
    }
  }
}

// ---------------------------------------------------------------------------
// Kernel 2: per-batch, find the 2 smallest-std token rows (ties -> later index,
// matching complement of jax top_k which keeps earlier index). Store sorted.
// ---------------------------------------------------------------------------
__global__ __launch_bounds__(256) void k_prune(const float* __restrict__ stdt,
                                               int* __restrict__ drop)
{
  __shared__ float vals[256];
  __shared__ int   idxs[256];
  const int b = blockIdx.x;
  const int tid = threadIdx.x;
  const float* sb = stdt + (size_t)b * (L_N - 1);
  int dropped0 = -1;
  int dfinal[2];
  for (int pass = 0; pass < 2; ++pass) {
    float bv = 3.402823e38f; int bi = -1;
    for (int i = tid; i < L_N - 1; i += 256) {
      if (i == dropped0) continue;
      float vv = sb[i];
      if (vv < bv || (vv == bv && i > bi)) { bv = vv; bi = i; }
    }
    vals[tid] = bv; idxs[tid] = bi;
    __syncthreads();
    for (int o = 128; o > 0; o >>= 1) {
      if (tid < o) {
        float vv = vals[tid + o]; int ii = idxs[tid + o];
        if (vv < vals[tid] || (vv == vals[tid] && ii > idxs[tid])) {
          vals[tid] = vv; idxs[tid] = ii;
        }
      }
      __syncthreads();
    }
    dfinal[pass] = idxs[0];
    dropped0 = idxs[0];
    __syncthreads();
  }
  if (tid == 0) {
    int lo = dfinal[0] < dfinal[1] ? dfinal[0] : dfinal[1];
    int hi = dfinal[0] < dfinal[1] ? dfinal[1] : dfinal[0];
    drop[b * 2 + 0] = lo;
    drop[b * 2 + 1] = hi;
  }
}

// ---------------------------------------------------------------------------
// Kernel 3: gather kept rows, 2nd softmax, write attn2 + idx3 (b128 streams),
// out = P @ V via split-K WMMA (8 waves = 4 N-tiles x 2 K-halves).
// Grid: (64, B), 256 threads. Ghost rows (>=1022 in last tile) are clamped so
// EXEC stays all-1s for WMMA; stores are masked instead.
// ---------------------------------------------------------------------------
__global__ __launch_bounds__(256) void k_gather_softmax_pv(
    const float* __restrict__ attn, const float* __restrict__ vmat,
    const int* __restrict__ drop,
    float* __restrict__ outp, float* __restrict__ attn2, int* __restrict__ idx3)
{
  __shared__ float ps[16 * SSTR];          // 16 x 1024 P strip; head reused for partials
  const int b    = blockIdx.y;
  const int t    = blockIdx.x;
  const int tid  = threadIdx.x;
  const int w    = tid >> 5;
  const int lane = tid & 31;
  const int d0 = drop[b * 2 + 0];
  const int d1 = drop[b * 2 + 1];

#pragma unroll
  for (int rr = 0; rr < 2; ++rr) {
    const int lr = w * 2 + rr;
    const int r  = t * 16 + lr;            // output row in [0, 1024)
    const int rc = (r < LOUT) ? r : (LOUT - 1);
    int kept = 0, src = 0;
    if (rc > 0) {                          // map compact index -> original token row
      int j = rc - 1;
      if (j >= d0) j++;
      if (j >= d1) j++;
      kept = j;
      src  = j + 1;                        // +1: attn_t starts at attn row 1
    }
    const float* arow = attn + ((size_t)b * L_N + src) * L_N;
    float* prow = &ps[lr * SSTR];
    float mx = -3.402823e38f;
    for (int j = lane * 4; j < L_N; j += 128) {
      v4f x = *(const v4f*)(arow + j);
      mx = fmaxf(mx, fmaxf(fmaxf(x.x, x.y), fmaxf(x.z, x.w)));
    }
    mx = wave_max32(mx);
    float s = 0.f;
    for (int j = lane * 4; j < L_N; j += 128) {
      v4f x = *(const v4f*)(arow + j);
      v4f e;
      e.x = __expf(x.x - mx); e.y = __expf(x.y - mx);
      e.z = __expf(x.z - mx); e.w = __expf(x.w - mx);
      *(v4f*)(prow + j) = e;
      s += (e.x + e.y) + (e.z + e.w);
    }
    s = wave_sum32(s);
    const float inv = 1.0f / s;
    const bool valid = (r < LOUT);
    float* a2row = attn2 + ((size_t)b * LOUT + rc) * L_N;
    int*   i3row = idx3 + ((size_t)b * NKEEP + (rc > 0 ? rc - 1 : 0)) * L_N;
    const v4i ki = {kept, kept, kept, kept};
    for (int j = lane * 4; j < L_N; j += 128) {
      v4f e = *(const v4f*)(prow + j);
      v4f p = e * inv;
      *(v4f*)(prow + j) = p;
      if (valid) *(v4f*)(a2row + j) = p;
      if (valid && r > 0) *(v4i*)(i3row + j) = ki;
    }
  }
  __syncthreads();

  // O[16x64] = P[16x1024] @ V[1024x64]; wave w: N-tile (w&3), K-half (w>>2)
  const int nt   = w & 3;
  const int half = w >> 2;
  const int m    = lane & 15;
  const int k0   = (lane < 16) ? 0 : 2;
  const float* vb = vmat + (size_t)b * L_N * D_N;
  v8f acc = {};
  const int kbase = half * 512;
  for (int blk = 0; blk < 8; ++blk) {      // 8 blocks x 64 keys
    if (blk < 7)                            // prefetch next V block (uniform branch)
      __builtin_prefetch(&vb[(size_t)(kbase + (blk + 1) * 64) * D_N + nt * 16 + m], 0, 1);
#pragma unroll
    for (int kc = 0; kc < 16; ++kc) {      // 64 keys / 4
      const int kk = kbase + blk * 64 + kc * 4 + k0;
      v2f af, bf;
      af.x = ps[m * SSTR + kk];
      af.y = ps[m * SSTR + kk + 1];
      bf.x = vb[(size_t)kk * D_N + nt * 16 + m];
      bf.y = vb[(size_t)(kk + 1) * D_N + nt * 16 + m];
      acc = __builtin_amdgcn_wmma_f32_16x16x4_f32(false, af, false, bf,
                                                  (short)0, acc, false, false);
    }
  }
  __syncthreads();                          // all waves done reading ps
  const int rbase = (lane < 16) ? 0 : 8;
  if (half == 1) {                          // stash upper-half partials in reused LDS
#pragma unroll
    for (int vv = 0; vv < 8; ++vv)
      ps[nt * 256 + (vv + rbase) * 16 + m] = acc[vv];
  }
  __syncthreads();
  if (half == 0) {
#pragma unroll
    for (int vv = 0; vv < 8; ++vv) {
      const int orow = t * 16 + vv + rbase;
      if (orow < LOUT) {
        float val = acc[vv] + ps[nt * 256 + (vv + rbase) * 16 + m];
        outp[((size_t)b * LOUT + orow) * D_N + nt * 16 + m] = val;
      }
    }
  }
}

// ---------------------------------------------------------------------------
extern "C" void kernel_launch(void* const* d_in, const int* in_sizes, int n_in,
                              void* d_out, int out_size, void* d_ws, size_t ws_size,
                              hipStream_t stream) {
  (void)in_sizes; (void)n_in; (void)out_size; (void)ws_size;
  const float* q = (const float*)d_in[0];
  const float* k = (const float*)d_in[1];
  const float* v = (const float*)d_in[2];

  // Workspace: attn (B*L*L f32) | std (B*1023 f32) | drop (B*2 i32)
  float* attn = (float*)d_ws;
  float* stdt = attn + (size_t)B_N * L_N * L_N;
  int*   drop = (int*)(stdt + (size_t)B_N * (L_N - 1));

  // Output: out (B*1022*64 f32) | attn2 (B*1022*1024 f32) | idx3 (B*1021*1024 i32)
  float* outp  = (float*)d_out;
  float* attn2 = outp + (size_t)B_N * LOUT * D_N;
  int*   idx3  = (int*)(attn2 + (size_t)B_N * LOUT * L_N);

  dim3 g1(L_N / 16, B_N);
  k_qk_softmax_std<<<g1, 256, 0, stream>>>(q, k, attn, stdt);
  k_prune<<<B_N, 256, 0, stream>>>(stdt, drop);
  dim3 g3(L_N / 16, B_N);
  k_gather_softmax_pv<<<g3, 256, 0, stream>>>(attn, v, drop, outp, attn2, idx3);
}